// TensorProduct_23373212025192
// MI455X (gfx1250) — compile-verified
//
#include <hip/hip_runtime.h>
#include <hip/hip_bf16.h>
#include <math.h>

// ---------------------------------------------------------------------------
// Problem geometry (fixed by the reference)
// ---------------------------------------------------------------------------
#define DIM1   16     // sum(2l+1, l=0..3) rows of x
#define DIM2   16     // rows of y
#define C_DIM  96     // channels
#define KOUT   99     // DIM_OUT
#define KPAD   112    // 7 tiles of 16 rows
#define NINSTR 23

// (l1, l2, l3, off1, off2, out_offset) for each instruction, in reference order
__constant__ int INSTR[NINSTR][6] = {
    {0,0,0, 0,0,  0}, {0,1,1, 0,1,  1}, {0,2,2, 0,4,  4}, {0,3,3, 0,9,  9},
    {1,0,1, 1,0, 16}, {1,1,0, 1,1, 19}, {1,1,2, 1,1, 20}, {1,2,1, 1,4, 25},
    {1,2,3, 1,4, 28}, {1,3,2, 1,9, 35},
    {2,0,2, 4,0, 40}, {2,1,1, 4,1, 45}, {2,1,3, 4,1, 48}, {2,2,0, 4,4, 55},
    {2,2,2, 4,4, 56}, {2,3,1, 4,9, 61}, {2,3,3, 4,9, 64},
    {3,0,3, 9,0, 71}, {3,1,2, 9,1, 78}, {3,2,1, 9,4, 83}, {3,2,3, 9,4, 86},
    {3,3,0, 9,9, 93}, {3,3,2, 9,9, 94},
};

// ---------------------------------------------------------------------------
// On-device Wigner-3j (real basis) coefficient generation  -> cgBig[99][16][16]
// ---------------------------------------------------------------------------
struct cplx { double re, im; };
__device__ inline cplx cmul(cplx a, cplx b) {
    return { a.re*b.re - a.im*b.im, a.re*b.im + a.im*b.re };
}
__device__ inline cplx cadd(cplx a, cplx b) { return { a.re + b.re, a.im + b.im }; }
__device__ inline cplx cconj(cplx a)        { return { a.re, -a.im }; }

__device__ inline double dfact(int n) {
    double r = 1.0;
    for (int i = 2; i <= n; ++i) r *= (double)i;
    return r;
}

__device__ double su2_cg(int j1, int j2, int j3, int m1, int m2, int m3) {
    if (m3 != m1 + m2) return 0.0;
    int vmin = max(max(-j1 + j2 + m3, -j1 + m1), 0);
    int vmax = min(min(j2 + j3 + m1, j3 - j1 + j2), j3 + m3);
    double pref = (2.0 * j3 + 1.0) *
        (dfact(j3 + j1 - j2) * dfact(j3 - j1 + j2) * dfact(j1 + j2 - j3) *
         dfact(j3 + m3) * dfact(j3 - m3)) /
        (dfact(j1 + j2 + j3 + 1) * dfact(j1 - m1) * dfact(j1 + m1) *
         dfact(j2 - m2) * dfact(j2 + m2));
    double s = 0.0;
    for (int v = vmin; v <= vmax; ++v) {
        double sign = ((v + j2 + m2) & 1) ? -1.0 : 1.0;
        s += sign * (dfact(j2 + j3 + m1 - v) * dfact(j1 - m1 + v)) /
             (dfact(v) * dfact(j3 - j1 + j2 - v) * dfact(j3 + m3 - v) *
              dfact(v + j1 - j2 - m3));
    }
    return sqrt(pref) * s;
}

__device__ void buildQ(int l, cplx Q[7][7]) {
    const double is2 = 0.70710678118654752440;
    int n = 2 * l + 1;
    for (int a = 0; a < 7; ++a)
        for (int b = 0; b < 7; ++b) Q[a][b] = { 0.0, 0.0 };
    for (int m = -l; m < 0; ++m) {
        Q[l + m][l - m] = { is2, 0.0 };   // column l+|m|
        Q[l + m][l + m] = { 0.0, -is2 };  // column l-|m|
    }
    Q[l][l] = { 1.0, 0.0 };
    for (int m = 1; m <= l; ++m) {
        double s = (m & 1) ? -1.0 : 1.0;
        Q[l + m][l + m] = { s * is2, 0.0 };
        Q[l + m][l - m] = { 0.0, s * is2 };
    }
    cplx p;
    switch (l & 3) {              // (-1j)^l
        case 0:  p = { 1.0, 0.0 };  break;
        case 1:  p = { 0.0, -1.0 }; break;
        case 2:  p = { -1.0, 0.0 }; break;
        default: p = { 0.0, 1.0 };  break;
    }
    for (int a = 0; a < n; ++a)
        for (int b = 0; b < n; ++b) Q[a][b] = cmul(p, Q[a][b]);
}

__global__ __launch_bounds__(256) void cg_init_kernel(float* __restrict__ cgBig) {
    int tid = threadIdx.x;
    for (int i = tid; i < KOUT * DIM1 * DIM2; i += blockDim.x) cgBig[i] = 0.0f;
    __syncthreads();
    if (tid >= NINSTR) return;

    int l1 = INSTR[tid][0], l2 = INSTR[tid][1], l3 = INSTR[tid][2];
    int o1 = INSTR[tid][3], o2 = INSTR[tid][4], ko = INSTR[tid][5];
    int n1 = 2 * l1 + 1, n2 = 2 * l2 + 1, n3 = 2 * l3 + 1;

    // Complex-basis CG tensor
    double Cc[7][7][7];
    for (int a = 0; a < n1; ++a)
        for (int b = 0; b < n2; ++b)
            for (int c = 0; c < n3; ++c) Cc[a][b][c] = 0.0;
    for (int m1 = -l1; m1 <= l1; ++m1)
        for (int m2 = -l2; m2 <= l2; ++m2) {
            int m3 = m1 + m2;
            if (m3 >= -l3 && m3 <= l3)
                Cc[l1 + m1][l2 + m2][l3 + m3] = su2_cg(l1, l2, l3, m1, m2, m3);
        }

    cplx Q1[7][7], Q2[7][7], Q3[7][7];
    buildQ(l1, Q1); buildQ(l2, Q2); buildQ(l3, Q3);

    // Staged basis change: Cr[j,l,m] = Re( sum_{i,k,n} Q1[i,j] Q2[k,l] conj(Q3[n,m]) Cc[i,k,n] )
    cplx T1[7][7][7];
    for (int i = 0; i < n1; ++i)
        for (int k = 0; k < n2; ++k)
            for (int m = 0; m < n3; ++m) {
                cplx acc = { 0.0, 0.0 };
                for (int n = 0; n < n3; ++n) {
                    cplx t = { Cc[i][k][n], 0.0 };
                    acc = cadd(acc, cmul(t, cconj(Q3[n][m])));
                }
                T1[i][k][m] = acc;
            }
    cplx T2[7][7][7];
    for (int i = 0; i < n1; ++i)
        for (int l = 0; l < n2; ++l)
            for (int m = 0; m < n3; ++m) {
                cplx acc = { 0.0, 0.0 };
                for (int k = 0; k < n2; ++k)
                    acc = cadd(acc, cmul(Q2[k][l], T1[i][k][m]));
                T2[i][l][m] = acc;
            }
    double Cr[7][7][7];
    double nrm2 = 0.0;
    for (int j = 0; j < n1; ++j)
        for (int l = 0; l < n2; ++l)
            for (int m = 0; m < n3; ++m) {
                cplx acc = { 0.0, 0.0 };
                for (int i = 0; i < n1; ++i)
                    acc = cadd(acc, cmul(Q1[i][j], T2[i][l][m]));
                Cr[j][l][m] = acc.re;
                nrm2 += acc.re * acc.re;
            }
    double scale = sqrt((double)(2 * l3 + 1)) / sqrt(nrm2);

    // Scatter: cgBig[ko+c][o1+a][o2+b] = cg[a,b,c]
    for (int a = 0; a < n1; ++a)
        for (int b = 0; b < n2; ++b)
            for (int c = 0; c < n3; ++c)
                cgBig[(ko + c) * (DIM1 * DIM2) + (o1 + a) * DIM2 + (o2 + b)] =
                    (float)(Cr[a][b][c] * scale);
}

// ---------------------------------------------------------------------------
// CDNA5 async global->LDS copy (ASYNCcnt-tracked, no VGPR round trip).
// On gfx1250 a flat shared-pointer carries the LDS byte offset in its low
// 32 bits (aperture lives in addr[63:32]), so truncation yields VDST.
// ---------------------------------------------------------------------------
__device__ __forceinline__ void async_copy_b128(void* lds_ptr, const void* gptr) {
    unsigned lds_off = (unsigned)(unsigned long long)lds_ptr;
    asm volatile("global_load_async_to_lds_b128 %0, %1, off"
                 :: "v"(lds_off), "v"(gptr) : "memory");
}
__device__ __forceinline__ void async_wait0() {
    asm volatile("s_wait_asynccnt 0" ::: "memory");
}

// ---------------------------------------------------------------------------
// Main tensor-product kernel: one workgroup (8 wave32) per edge.
//   M[e] (99x16)  = cgBig . y[e]        (built in LDS, VALU)
//   out[e](99x96) = M[e] @ x[e](16x96)  (V_WMMA_F32_16X16X4_F32, K=16 in 4 steps)
// ---------------------------------------------------------------------------
typedef __attribute__((ext_vector_type(2))) float v2f;
typedef __attribute__((ext_vector_type(8))) float v8f;

// One 16x16 output tile: K=16 contraction as 4 chained WMMA(K=4) ops.
__device__ __forceinline__ v8f tile_mma(const float* sX, const v2f A[4],
                                        int half, int col) {
    v8f acc = {};
    #pragma unroll
    for (int kc = 0; kc < 4; ++kc) {
        v2f B;
        const int krow = kc * 4 + half;
        B.x = sX[(krow + 0) * C_DIM + col];
        B.y = sX[(krow + 1) * C_DIM + col];
        acc = __builtin_amdgcn_wmma_f32_16x16x4_f32(
                false, A[kc], false, B, (short)0, acc, false, false);
    }
    return acc;
}

__global__ __launch_bounds__(256) void tp_wmma_kernel(
        const float* __restrict__ x, const float* __restrict__ y,
        const float* __restrict__ cgBig, float* __restrict__ out) {
    __shared__ float sM[KPAD * DIM1];   // 112x16, rows >= 99 zero
    __shared__ float sX[DIM1 * C_DIM];  // 16x96
    __shared__ float sY[DIM2];

    const int e   = blockIdx.x;
    const int tid = threadIdx.x;

    // ---- async-stage x (6 KB) and y (64 B) into LDS -----------------------
    const float* xe = x + (size_t)e * (DIM1 * C_DIM);
    const float* ye = y + (size_t)e * DIM2;
    for (int i = tid; i < (DIM1 * C_DIM) / 4; i += 256)       // 384 x b128
        async_copy_b128(&sX[i * 4], xe + i * 4);
    if (tid < DIM2 / 4)                                       // 4 x b128
        async_copy_b128(&sY[tid * 4], ye + tid * 4);
    async_wait0();
    __syncthreads();

    // ---- build M[k][i] = sum_j cgBig[k][i][j] * y[j] ----------------------
    for (int idx = tid; idx < KPAD * DIM1; idx += 256) {
        float acc = 0.0f;
        int k = idx >> 4;
        if (k < KOUT) {
            const float* cgr = cgBig + (size_t)idx * DIM2; // (k*16+i)*16
            #pragma unroll
            for (int j = 0; j < DIM2; ++j) acc += cgr[j] * sY[j];
        }
        sM[idx] = acc;
    }
    __syncthreads();

    // ---- WMMA: wave w owns output rows 16w..16w+15 ------------------------
    const int wave = tid >> 5;
    const int lane = tid & 31;
    if (wave >= 7) return;        // wave 7: load/M-build helper only

    const int half = (lane < 16) ? 0 : 2;   // A/B K-pair select per ISA layout
    const int l15  = lane & 15;
    const int arow = wave * 16 + l15;

    // A operand per k-chunk: lanes<16 -> {K=4kc, 4kc+1}, lanes>=16 -> {4kc+2, 4kc+3}
    v2f A[4];
    #pragma unroll
    for (int kc = 0; kc < 4; ++kc) {
        A[kc].x = sM[arow * DIM1 + kc * 4 + half + 0];
        A[kc].y = sM[arow * DIM1 + kc * 4 + half + 1];
    }

    float* oe = out + (size_t)e * (KOUT * C_DIM);
    const int rbase = wave * 16 + ((lane < 16) ? 0 : 8);  // C/D row layout

    if (wave < 6) {
        // Full tiles: rows all valid -> unconditional streaming (NT) stores.
        #pragma unroll
        for (int n = 0; n < C_DIM / 16; ++n) {
            const int col = n * 16 + l15;
            v8f acc = tile_mma(sX, A, half, col);
            float* p = oe + (size_t)rbase * C_DIM + col;
            #pragma unroll
            for (int v = 0; v < 8; ++v)
                __builtin_nontemporal_store(acc[v], p + (size_t)v * C_DIM);
        }
    } else {
        // Boundary tile (rows 96..111): only rows 96..98 exist (lanes < 16, v < 3).
        #pragma unroll
        for (int n = 0; n < C_DIM / 16; ++n) {
            const int col = n * 16 + l15;
            v8f acc = tile_mma(sX, A, half, col);
            if (lane < 16) {
                float* p = oe + (size_t)96 * C_DIM + col;
                __builtin_nontemporal_store(acc[0], p + 0 * C_DIM);
                __builtin_nontemporal_store(acc[1], p + 1 * C_DIM);
                __builtin_nontemporal_store(acc[2], p + 2 * C_DIM);
            }
        }
    }
}

// ---------------------------------------------------------------------------
extern "C" void kernel_launch(void* const* d_in, const int* in_sizes, int n_in,
                              void* d_out, int out_size, void* d_ws, size_t ws_size,
                              hipStream_t stream) {
    const float* x = (const float*)d_in[0];
    const float* y = (const float*)d_in[1];
    float* out     = (float*)d_out;
    float* cgBig   = (float*)d_ws;   // 99*16*16 floats = 101376 B

    const int E = in_sizes[0] / (DIM1 * C_DIM);

    cg_init_kernel<<<1, 256, 0, stream>>>(cgBig);
    tp_wmma_kernel<<<E, 256, 0, stream>>>(x, y, cgBig, out);
}